// SURM_Module_80942953660659
// MI455X (gfx1250) — compile-verified
//
#include <hip/hip_runtime.h>
#include <hip/hip_bf16.h>
#include <cstdint>

// ---------------------------------------------------------------- constants
#define BB 16
#define CC 96
#define PP 16384            // H*W
#define KPI 31              // top-k per image
#define KTOT (BB*KPI)       // 496
#define HSC 64              // scoring hidden
#define HDC 128             // decoder hidden
#define SAPITCH 98          // LDS tile pitch (even -> 8B-aligned ds_load_b64)

// d_out layout (floats): opt_tok | sar_upd | recon_loss | total_kl | unc_map
#define OUT_OPT_TOK 0
#define OUT_SAR_UPD ((size_t)BB*PP*CC)                  // 25165824
#define OUT_RECON   (2*(size_t)BB*PP*CC)                // 50331648
#define OUT_KL      (2*(size_t)BB*PP*CC + 1)
#define OUT_UNC     (2*(size_t)BB*PP*CC + 2)

typedef __attribute__((ext_vector_type(2))) float v2f;
typedef __attribute__((ext_vector_type(8))) float v8f;

// ---------------------------------------------------------------- helpers
__device__ __forceinline__ float redsum16(float x) {
  x += __shfl_xor(x, 1, 32);
  x += __shfl_xor(x, 2, 32);
  x += __shfl_xor(x, 4, 32);
  x += __shfl_xor(x, 8, 32);
  return x;
}
__device__ __forceinline__ float redmax16(float x) {
  x = fmaxf(x, __shfl_xor(x, 1, 32));
  x = fmaxf(x, __shfl_xor(x, 2, 32));
  x = fmaxf(x, __shfl_xor(x, 4, 32));
  x = fmaxf(x, __shfl_xor(x, 8, 32));
  return x;
}
__device__ __forceinline__ float redsum32(float x) {
  x = redsum16(x);
  x += __shfl_xor(x, 16, 32);
  return x;
}

// Stage x[c][p0+i] (c=0..95, i=0..15) into per-wave LDS tile sA[i*SAPITCH + c].
__device__ __forceinline__ void stage_tile(const float* __restrict__ xg,
                                           int p0, int lane, float* sA) {
#pragma unroll 4
  for (int it = 0; it < 48; ++it) {
    const int flat = it * 32 + lane;
    const int c = flat >> 4;
    const int i = flat & 15;
    sA[i * SAPITCH + c] = xg[(size_t)c * PP + p0 + i];
  }
}

// LDS tile x W[96x96] -> 16x96 via 6 WMMA C/D tiles; ks-outer real loop keeps
// register pressure low (A frag = one ds_load_b64 per iteration).
__device__ __forceinline__ void gemm96(const float* sA,
                                       const float* __restrict__ W,
                                       const float* __restrict__ bias,
                                       int m, int hi, v8f* out) {
  const v8f zero = {0.f, 0.f, 0.f, 0.f, 0.f, 0.f, 0.f, 0.f};
#pragma unroll
  for (int t = 0; t < 6; ++t) out[t] = zero;
#pragma unroll 2
  for (int ks = 0; ks < 24; ++ks) {
    const int k = 4 * ks + 2 * hi;
    v2f a;
    a.x = sA[m * SAPITCH + k];
    a.y = sA[m * SAPITCH + k + 1];
    const float* Wr0 = W + k * CC + m;
    v2f bf[6];
#pragma unroll
    for (int t = 0; t < 6; ++t) {
      bf[t].x = Wr0[t * 16];
      bf[t].y = Wr0[CC + t * 16];
    }
#pragma unroll
    for (int t = 0; t < 6; ++t)
      out[t] = __builtin_amdgcn_wmma_f32_16x16x4_f32(
          false, a, false, bf[t], (short)0, out[t], false, false);
  }
#pragma unroll
  for (int t = 0; t < 6; ++t) {
    const float bv = bias[t * 16 + m];
#pragma unroll
    for (int j = 0; j < 8; ++j) out[t][j] += bv;
  }
}

// ---------------------------------------------------------------- kernels
__global__ void init_kernel(float* acc) {
  if (threadIdx.x < 3) acc[threadIdx.x] = 0.f;
}

// [B][C][P] -> [B][P][C], LDS tiled
__global__ __launch_bounds__(256) void transpose_kernel(
    const float* __restrict__ in, float* __restrict__ out) {
  __shared__ float tile[32][33];
  const int b = blockIdx.z, c0 = blockIdx.y * 32, p0 = blockIdx.x * 32;
  const int tx = threadIdx.x, ty = threadIdx.y;  // 32 x 8
  const float* ib = in + (size_t)b * CC * PP;
  float* ob = out + (size_t)b * PP * CC;
#pragma unroll
  for (int i = 0; i < 4; ++i)
    tile[ty + i * 8][tx] = ib[(size_t)(c0 + ty + i * 8) * PP + p0 + tx];
  __syncthreads();
#pragma unroll
  for (int i = 0; i < 4; ++i)
    ob[(size_t)(p0 + ty + i * 8) * CC + c0 + tx] = tile[tx][ty + i * 8];
}

// Fused: 4 GEMMs (WMMA fp32) + per-patch v + align_kl partial. 1 wave = 16 patches.
__global__ __launch_bounds__(64) void encode_kernel(
    const float* __restrict__ optx, const float* __restrict__ sarx,
    const float* __restrict__ mu_w, const float* __restrict__ mu_b,
    const float* __restrict__ lv_w, const float* __restrict__ lv_b,
    float* __restrict__ v_out, float* __restrict__ align_acc) {
  __shared__ float sAall[2][16 * SAPITCH];
  const int wid = threadIdx.x >> 5;
  const int lane = threadIdx.x & 31;
  const int m = lane & 15, hi = lane >> 4;
  const int wave = blockIdx.x * 2 + wid;
  const int b = wave / (PP / 16);
  const int p0 = (wave % (PP / 16)) * 16;
  float* sA = sAall[wid];

  const float* xo = optx + (size_t)b * CC * PP;
  const float* xs = sarx + (size_t)b * CC * PP;

  // ---- opt phase
  stage_tile(xo, p0, lane, sA);
  __syncthreads();

  // opt_logvar -> per-patch channel sums (accumulators freed immediately)
  float slv_opt[8];
  {
    v8f lv[6];
    gemm96(sA, lv_w, lv_b, m, hi, lv);
#pragma unroll
    for (int j = 0; j < 8; ++j) {
      float s = 0.f;
#pragma unroll
      for (int t = 0; t < 6; ++t) s += lv[t][j];
      slv_opt[j] = redsum16(s);
    }
  }

  // opt_mu tile -> transform in place to lt = log_softmax(opt_mu)
  v8f lt[6];
  gemm96(sA, mu_w, mu_b, m, hi, lt);
#pragma unroll
  for (int j = 0; j < 8; ++j) {
    float mx = lt[0][j];
#pragma unroll
    for (int t = 1; t < 6; ++t) mx = fmaxf(mx, lt[t][j]);
    mx = redmax16(mx);
    float se = 0.f;
#pragma unroll
    for (int t = 0; t < 6; ++t) se += expf(lt[t][j] - mx);
    const float off = mx + logf(redsum16(se));
#pragma unroll
    for (int t = 0; t < 6; ++t) lt[t][j] -= off;
  }

  // ---- sar phase (re-stage tile; same LDS region)
  __syncthreads();
  stage_tile(xs, p0, lane, sA);
  __syncthreads();

  // sar_mu + align_kl:  sum t*(logt - logp), t = exp(lt)
  float aacc = 0.f;
  {
    v8f mus[6];
    gemm96(sA, mu_w, mu_b, m, hi, mus);
#pragma unroll
    for (int j = 0; j < 8; ++j) {
      float smx = mus[0][j];
#pragma unroll
      for (int t = 1; t < 6; ++t) smx = fmaxf(smx, mus[t][j]);
      smx = redmax16(smx);
      float sse = 0.f;
#pragma unroll
      for (int t = 0; t < 6; ++t) sse += expf(mus[t][j] - smx);
      const float soff = smx + logf(redsum16(sse));
#pragma unroll
      for (int t = 0; t < 6; ++t) {
        const float l = lt[t][j];
        const float lp = mus[t][j] - soff;
        aacc += expf(l) * (l - lp);
      }
    }
  }
  aacc = redsum32(aacc);
  if (lane == 0) atomicAdd(align_acc, aacc);

  // sar_logvar (clipped) -> sums -> v
  float slv_sar[8];
  {
    v8f lv[6];
    gemm96(sA, lv_w, lv_b, m, hi, lv);
#pragma unroll
    for (int j = 0; j < 8; ++j) {
      float s = 0.f;
#pragma unroll
      for (int t = 0; t < 6; ++t)
        s += fminf(fmaxf(lv[t][j], -10.f), 10.f);
      slv_sar[j] = redsum16(s);
    }
  }
#pragma unroll
  for (int j = 0; j < 8; ++j) {
    const float vv = 0.5f * (logf(expf(slv_sar[j]) + 1e-6f) -
                             logf(expf(slv_opt[j]) + 1e-6f));
    if (m == j) v_out[(size_t)b * PP + p0 + 8 * hi + j] = vv;
  }
}

// scoring MLP + softmax; one block per image
__global__ __launch_bounds__(256) void score_kernel(
    const float* __restrict__ v, const float* __restrict__ fc1_w,
    const float* __restrict__ fc1_b, const float* __restrict__ fc2_w,
    const float* __restrict__ fc2_b, float* __restrict__ scores,
    float* __restrict__ unc_out) {
  __shared__ float red[256];
  __shared__ float hsm[HSC];
  const int b = blockIdx.x, tid = threadIdx.x;
  const float* vb = v + (size_t)b * PP;

  const int j = tid & 63, seg = tid >> 6;
  float ha = 0.f;
  for (int p = seg; p < PP; p += 4) ha += vb[p] * fc1_w[(size_t)p * HSC + j];
  red[tid] = ha;
  __syncthreads();
  if (tid < HSC)
    hsm[tid] = fmaxf(red[tid] + red[tid + 64] + red[tid + 128] + red[tid + 192] +
                         fc1_b[tid], 0.f);
  __syncthreads();

  float raw[64];
  float lmax = -3.4e38f;
#pragma unroll 4
  for (int i = 0; i < 64; ++i) {
    const int p = tid + (i << 8);
    float r = fc2_b[p];
#pragma unroll
    for (int jj = 0; jj < HSC; ++jj) r += hsm[jj] * fc2_w[(size_t)jj * PP + p];
    raw[i] = r;
    lmax = fmaxf(lmax, r);
  }
  red[tid] = lmax;
  __syncthreads();
  for (int s = 128; s > 0; s >>= 1) {
    if (tid < s) red[tid] = fmaxf(red[tid], red[tid + s]);
    __syncthreads();
  }
  const float bmax = red[0];
  __syncthreads();
  float lsum = 0.f;
#pragma unroll 4
  for (int i = 0; i < 64; ++i) lsum += expf(raw[i] - bmax);
  red[tid] = lsum;
  __syncthreads();
  for (int s = 128; s > 0; s >>= 1) {
    if (tid < s) red[tid] += red[tid + s];
    __syncthreads();
  }
  const float inv = 1.f / red[0];
#pragma unroll 4
  for (int i = 0; i < 64; ++i) {
    const int p = tid + (i << 8);
    const float sc = expf(raw[i] - bmax) * inv;
    scores[(size_t)b * PP + p] = sc;
    unc_out[(size_t)b * PP + p] = sc;
  }
}

// per-image top-31, descending value, smaller index wins ties (matches lax.top_k)
__global__ __launch_bounds__(256) void topk_kernel(float* __restrict__ scores,
                                                   int* __restrict__ topk) {
  __shared__ unsigned long long red[256];
  const int b = blockIdx.x, tid = threadIdx.x;
  float* sb = scores + (size_t)b * PP;
  for (int kk = 0; kk < KPI; ++kk) {
    unsigned long long best = 0ull;
    for (int p = tid; p < PP; p += 256) {
      const float val = sb[p];
      if (val >= 0.f) {
        const unsigned long long key =
            ((unsigned long long)__float_as_uint(val) << 32) |
            (unsigned)(PP - 1 - p);
        if (key > best) best = key;
      }
    }
    red[tid] = best;
    __syncthreads();
    for (int s = 128; s > 0; s >>= 1) {
      if (tid < s && red[tid + s] > red[tid]) red[tid] = red[tid + s];
      __syncthreads();
    }
    const unsigned long long w = red[0];
    __syncthreads();
    const int p = (PP - 1) - (int)(w & 0xffffffffu);
    if (tid == 0) {
      topk[b * KPI + kk] = p;
      sb[p] = -1.f;  // exclude from further rounds
    }
    __syncthreads();
  }
}

// gather + recompute mu/lv + reparam + decoder + scatter + loss partials
__global__ __launch_bounds__(128) void gather_dec_kernel(
    const float* __restrict__ optx, const float* __restrict__ sarx,
    const float* __restrict__ mu_w, const float* __restrict__ mu_b,
    const float* __restrict__ lv_w, const float* __restrict__ lv_b,
    const float* __restrict__ d1w, const float* __restrict__ d1b,
    const float* __restrict__ d2w, const float* __restrict__ d2b,
    const float* __restrict__ eps, const int* __restrict__ topk,
    float* __restrict__ sar_upd, float* __restrict__ acc_kl,
    float* __restrict__ acc_rec) {
  __shared__ float xf[CC], zz[CC], hh[HDC], red[128];
  const int k = blockIdx.x, b = k / KPI, tid = threadIdx.x;
  const int p = topk[k];

  float oldv = 0.f;
  if (tid < CC) {
    xf[tid] = optx[((size_t)b * CC + tid) * PP + p];
    oldv = sarx[((size_t)b * CC + tid) * PP + p];
  }
  __syncthreads();

  float klp = 0.f;
  if (tid < CC) {
    float m_ = mu_b[tid], l_ = lv_b[tid];
#pragma unroll 8
    for (int jj = 0; jj < CC; ++jj) {
      const float xv = xf[jj];
      m_ += xv * mu_w[jj * CC + tid];
      l_ += xv * lv_w[jj * CC + tid];
    }
    zz[tid] = m_ + expf(0.5f * l_) * eps[(size_t)k * CC + tid];
    klp = 1.f + l_ - m_ * m_ - expf(l_);
  }
  red[tid] = klp;
  __syncthreads();
  for (int s = 64; s > 0; s >>= 1) {
    if (tid < s) red[tid] += red[tid + s];
    __syncthreads();
  }
  if (tid == 0) atomicAdd(acc_kl, red[0]);
  __syncthreads();

  if (tid < HDC) {
    float h_ = d1b[tid];
#pragma unroll 8
    for (int jj = 0; jj < CC; ++jj) h_ += zz[jj] * d1w[jj * HDC + tid];
    hh[tid] = fmaxf(h_, 0.f);
  }
  __syncthreads();

  float rp = 0.f;
  if (tid < CC) {
    float r_ = d2b[tid];
#pragma unroll 8
    for (int jj = 0; jj < HDC; ++jj) r_ += hh[jj] * d2w[jj * CC + tid];
    const float upd = 0.5f * r_ + 0.5f * oldv;  // (1-GAMMA)*recon + GAMMA*old
    sar_upd[((size_t)b * PP + p) * CC + tid] = upd;
    const float d = upd - xf[tid];
    rp = d * d;
  }
  red[tid] = rp;
  __syncthreads();
  for (int s = 64; s > 0; s >>= 1) {
    if (tid < s) red[tid] += red[tid + s];
    __syncthreads();
  }
  if (tid == 0) atomicAdd(acc_rec, red[0]);
}

__global__ void fin_kernel(const float* __restrict__ acc,
                           float* __restrict__ out_rec,
                           float* __restrict__ out_kl) {
  if (threadIdx.x == 0) {
    *out_rec = acc[2] / (float)(KTOT * CC);
    *out_kl = -0.5f * acc[1] / (float)BB + acc[0] / (float)BB;
  }
}

// ---------------------------------------------------------------- launch
extern "C" void kernel_launch(void* const* d_in, const int* in_sizes, int n_in,
                              void* d_out, int out_size, void* d_ws,
                              size_t ws_size, hipStream_t stream) {
  const float* opt_tok = (const float*)d_in[0];
  const float* sar_tok = (const float*)d_in[1];
  const float* mu_w = (const float*)d_in[2];
  const float* mu_b = (const float*)d_in[3];
  const float* lv_w = (const float*)d_in[4];
  const float* lv_b = (const float*)d_in[5];
  const float* dec1_w = (const float*)d_in[6];
  const float* dec1_b = (const float*)d_in[7];
  const float* dec2_w = (const float*)d_in[8];
  const float* dec2_b = (const float*)d_in[9];
  const float* fc1_w = (const float*)d_in[10];
  const float* fc1_b = (const float*)d_in[11];
  const float* fc2_w = (const float*)d_in[12];
  const float* fc2_b = (const float*)d_in[13];
  const float* eps = (const float*)d_in[14];

  float* out = (float*)d_out;
  float* out_opt = out + OUT_OPT_TOK;
  float* out_sar = out + OUT_SAR_UPD;
  float* out_unc = out + OUT_UNC;

  float* ws = (float*)d_ws;
  float* v_buf = ws;                       // B*P
  float* scores = ws + (size_t)BB * PP;    // B*P
  float* acc = ws + 2 * (size_t)BB * PP;   // [align, kl, rec]
  int* topk = (int*)(ws + 2 * (size_t)BB * PP + 4);

  init_kernel<<<1, 32, 0, stream>>>(acc);

  dim3 tb(32, 8), tg(PP / 32, CC / 32, BB);
  transpose_kernel<<<tg, tb, 0, stream>>>(opt_tok, out_opt);
  transpose_kernel<<<tg, tb, 0, stream>>>(sar_tok, out_sar);

  encode_kernel<<<(BB * PP / 16) / 2, 64, 0, stream>>>(
      opt_tok, sar_tok, mu_w, mu_b, lv_w, lv_b, v_buf, acc + 0);

  score_kernel<<<BB, 256, 0, stream>>>(v_buf, fc1_w, fc1_b, fc2_w, fc2_b,
                                       scores, out_unc);

  topk_kernel<<<BB, 256, 0, stream>>>(scores, topk);

  gather_dec_kernel<<<KTOT, 128, 0, stream>>>(
      opt_tok, sar_tok, mu_w, mu_b, lv_w, lv_b, dec1_w, dec1_b, dec2_w, dec2_b,
      eps, topk, out_sar, acc + 1, acc + 2);

  fin_kernel<<<1, 32, 0, stream>>>(acc, out + OUT_RECON, out + OUT_KL);
}